// M2_MOEP_22351009809019
// MI455X (gfx1250) — compile-verified
//
#include <hip/hip_runtime.h>
#include <math.h>
#include <stdint.h>

// ---------------------------------------------------------------------------
// dims: B=32, L=512, D=32, H=256, E=8, PLEN=96, O=32, LAT=256, PD=64, NPROT=16
// ---------------------------------------------------------------------------

typedef __attribute__((ext_vector_type(2))) float v2f;
typedef __attribute__((ext_vector_type(8))) float v8f;
typedef __attribute__((ext_vector_type(4))) unsigned int u32x4;
typedef __attribute__((ext_vector_type(4))) int i32x4;
typedef __attribute__((ext_vector_type(8))) int i32x8;

// ---------------------------------------------------------------------------
// fp32 WMMA 16x16 tile GEMM, A from global: C += A(16xK) * B(Kx16), K step 4.
// A layout (ISA 32-bit A 16x4): lane m = lane%16; K split across lane halves.
// C/D layout: VGPR r -> (M = r + hi*8, N = lane%16).
// ---------------------------------------------------------------------------
__device__ __forceinline__ v8f gemm_tile_f32(const float* __restrict__ A, int lda,
                                             const float* __restrict__ Bm, int ldb,
                                             int K, int m0, int n0) {
  const int lane = threadIdx.x & 31;
  const int lm   = lane & 15;
  const int hi   = lane >> 4;
  const int k0   = hi * 2;
  v8f c = {};
  const float* arow = A + (size_t)(m0 + lm) * (size_t)lda;
  for (int k = 0; k < K; k += 4) {
    v2f a, b;
    a.x = arow[k + k0];
    a.y = arow[k + k0 + 1];
    b.x = Bm[(size_t)(k + k0)     * (size_t)ldb + n0 + lm];
    b.y = Bm[(size_t)(k + k0 + 1) * (size_t)ldb + n0 + lm];
    c = __builtin_amdgcn_wmma_f32_16x16x4_f32(false, a, false, b, (short)0, c,
                                              false, false);
  }
  return c;
}

// Same, but A fragments come from an LDS-staged tile (padded row stride so the
// 16 lanes of the A fragment hit 16 distinct banks: lane*ldas mod 64 distinct).
__device__ __forceinline__ v8f gemm_tile_f32_ldsA(const float* As, int ldas,
                                                  const float* __restrict__ Bm,
                                                  int ldb, int K, int n0) {
  const int lane = threadIdx.x & 31;
  const int lm   = lane & 15;
  const int hi   = lane >> 4;
  const int k0   = hi * 2;
  v8f c = {};
  const float* arow = As + lm * ldas;
  for (int k = 0; k < K; k += 4) {
    v2f a, b;
    a.x = arow[k + k0];
    a.y = arow[k + k0 + 1];
    b.x = Bm[(size_t)(k + k0)     * (size_t)ldb + n0 + lm];
    b.y = Bm[(size_t)(k + k0 + 1) * (size_t)ldb + n0 + lm];
    c = __builtin_amdgcn_wmma_f32_16x16x4_f32(false, a, false, b, (short)0, c,
                                              false, false);
  }
  return c;
}

// ---------------------------------------------------------------------------
// 512-point radix-2 DIT FFT in LDS, 256 threads (1 butterfly/thread/stage).
// ---------------------------------------------------------------------------
__device__ __forceinline__ void fft512(float* re, float* im, int t) {
  for (int s = 1; s <= 9; ++s) {
    const int half = 1 << (s - 1);
    const int len  = half << 1;
    const int blk  = t / half;
    const int pos  = t - blk * half;
    const int i0   = blk * len + pos;
    const int i1   = i0 + half;
    float c, sn;
    __sincosf(-6.283185307179586f * (float)pos / (float)len, &sn, &c);
    __syncthreads();
    const float xr = re[i1], xi = im[i1];
    const float tr = xr * c - xi * sn;
    const float ti = xr * sn + xi * c;
    const float ur = re[i0], ui = im[i0];
    re[i0] = ur + tr; im[i0] = ui + ti;
    re[i1] = ur - tr; im[i1] = ui - ti;
  }
  __syncthreads();
}

// ---------------------------------------------------------------------------
// K0: per-(b,d) stats of x[b,:,d]. grid = 1024, block = 256
// ---------------------------------------------------------------------------
__global__ void k_stats(const float* __restrict__ x, float* __restrict__ cf) {
  __shared__ float xs[512], re[512], im[512], red[256];
  const int t = threadIdx.x;
  const int bd = blockIdx.x;
  const int b = bd >> 5, d = bd & 31;
  const float* base = x + (size_t)b * 512 * 32 + d;
  const float v0 = base[(size_t)t * 32];
  const float v1 = base[(size_t)(t + 256) * 32];
  xs[t] = v0; xs[t + 256] = v1;

  red[t] = v0 + v1; __syncthreads();
  for (int s = 128; s > 0; s >>= 1) { if (t < s) red[t] += red[t + s]; __syncthreads(); }
  const float sum = red[0]; __syncthreads();
  red[t] = v0 * v0 + v1 * v1; __syncthreads();
  for (int s = 128; s > 0; s >>= 1) { if (t < s) red[t] += red[t + s]; __syncthreads(); }
  const float sumsq = red[0]; __syncthreads();
  red[t] = fminf(v0, v1); __syncthreads();
  for (int s = 128; s > 0; s >>= 1) { if (t < s) red[t] = fminf(red[t], red[t + s]); __syncthreads(); }
  const float mn = red[0]; __syncthreads();
  red[t] = fmaxf(v0, v1); __syncthreads();
  for (int s = 128; s > 0; s >>= 1) { if (t < s) red[t] = fmaxf(red[t], red[t + s]); __syncthreads(); }
  const float mx = red[0]; __syncthreads();

  {
    const int j0 = (int)(__brev((unsigned)t) >> 23);
    const int j1 = (int)(__brev((unsigned)(t + 256)) >> 23);
    re[j0] = v0; im[j0] = 0.f;
    re[j1] = v1; im[j1] = 0.f;
  }
  fft512(re, im, t);
  const float m0 = sqrtf(re[t] * re[t] + im[t] * im[t]);
  const float m1 = sqrtf(re[t + 256] * re[t + 256] + im[t + 256] * im[t + 256]);
  red[t] = m0 + m1; __syncthreads();
  for (int s = 128; s > 0; s >>= 1) { if (t < s) red[t] += red[t + s]; __syncthreads(); }
  const float fsum = red[0];

  if (t == 0) {
    const float mean = sum / 512.f;
    const float var  = (sumsq - sum * sum / 512.f) / 511.f;   // ddof=1
    const float sd   = sqrtf(fmaxf(var, 0.f));
    const float x0 = xs[0], xl = xs[511];
    float* row = cf + b * 224;
    row[0 * 32 + d] = mean;
    row[1 * 32 + d] = sd;
    row[2 * 32 + d] = mn;
    row[3 * 32 + d] = mx;
    row[4 * 32 + d] = (xl - x0) / 511.f;
    row[5 * 32 + d] = (xl - x0) / 512.f;
    row[6 * 32 + d] = fsum / 512.f;
  }
}

// ---------------------------------------------------------------------------
// K1: expert-feature MLP + LayerNorm (tiny). grid=1, block=256
// ---------------------------------------------------------------------------
__global__ void k_efmlp(const float* __restrict__ cf,
                        const float* __restrict__ Wf1, const float* __restrict__ bf1,
                        const float* __restrict__ Wf2, const float* __restrict__ bf2,
                        const float* __restrict__ lng, const float* __restrict__ lnb,
                        float* __restrict__ h1, float* __restrict__ efpre,
                        float* __restrict__ ef) {
  const int t = threadIdx.x;
  for (int i = t; i < 32 * 256; i += 256) {
    const int b = i >> 8, j = i & 255;
    float acc = bf1[j];
    const float* c = cf + b * 224;
    for (int k = 0; k < 224; ++k) acc += c[k] * Wf1[k * 256 + j];
    h1[i] = fmaxf(acc, 0.f);
  }
  __syncthreads();
  for (int i = t; i < 32 * 64; i += 256) {
    const int b = i >> 6, j = i & 63;
    float acc = bf2[j];
    const float* hh = h1 + b * 256;
    for (int k = 0; k < 256; ++k) acc += hh[k] * Wf2[k * 64 + j];
    efpre[i] = acc;
  }
  __syncthreads();
  if (t < 32) {
    const float* p = efpre + t * 64;
    float mu = 0.f;
    for (int j = 0; j < 64; ++j) mu += p[j];
    mu /= 64.f;
    float var = 0.f;
    for (int j = 0; j < 64; ++j) { const float dd = p[j] - mu; var += dd * dd; }
    var /= 64.f;
    const float inv = rsqrtf(var + 1e-5f);
    for (int j = 0; j < 64; ++j) ef[t * 64 + j] = (p[j] - mu) * inv * lng[j] + lnb[j];
  }
}

// ---------------------------------------------------------------------------
// K2: hidden = relu(xflat[32,16384] @ W1[16384,256] + b1)  (WMMA f32, direct A:
// the 1 MB A-tile cannot be LDS-staged; kernel is bandwidth-trivial anyway)
// ---------------------------------------------------------------------------
__global__ void k_latgemm(const float* __restrict__ x, const float* __restrict__ W1,
                          const float* __restrict__ b1, float* __restrict__ hidden) {
  const int wid = blockIdx.x * 8 + (threadIdx.x >> 5);
  const int m0 = (wid >> 4) * 16, n0 = (wid & 15) * 16;
  const v8f c = gemm_tile_f32(x, 16384, W1, 256, 16384, m0, n0);
  const int lane = threadIdx.x & 31, lm = lane & 15, hi = lane >> 4;
  for (int r = 0; r < 8; ++r) {
    const int row = m0 + r + hi * 8, col = n0 + lm;
    hidden[row * 256 + col] = fmaxf(c[r] + b1[col], 0.f);
  }
}

// ---------------------------------------------------------------------------
// K3: latent = hidden@W2+b2 ; lg = latent@Wg+bg (tiny). grid=1, block=256
// ---------------------------------------------------------------------------
__global__ void k_latent2(const float* __restrict__ hidden,
                          const float* __restrict__ W2, const float* __restrict__ b2,
                          const float* __restrict__ Wg, const float* __restrict__ bg,
                          float* __restrict__ latent, float* __restrict__ lg) {
  const int t = threadIdx.x;
  for (int i = t; i < 32 * 256; i += 256) {
    const int b = i >> 8, j = i & 255;
    float acc = b2[j];
    const float* hh = hidden + b * 256;
    for (int k = 0; k < 256; ++k) acc += hh[k] * W2[k * 256 + j];
    latent[i] = acc;
  }
  __syncthreads();
  {
    const int b = t >> 3, e = t & 7;   // 32*8 == 256 entries exactly
    float acc = bg[e];
    const float* la = latent + b * 256;
    for (int k = 0; k < 256; ++k) acc += la[k] * Wg[k * 8 + e];
    lg[t] = acc;
  }
}

// ---------------------------------------------------------------------------
// K4: prototype projection, distances, softmax gate. grid=1, block=256
// ---------------------------------------------------------------------------
__global__ void k_gate(const float* __restrict__ ef,
                       const float* __restrict__ Wp, const float* __restrict__ bp,
                       const float* __restrict__ prot, const float* __restrict__ lg,
                       const float* __restrict__ log_temp, float* __restrict__ gw) {
  __shared__ float proj[32 * 64];
  __shared__ float d2s[32 * 16];
  const int t = threadIdx.x;
  for (int i = t; i < 2048; i += 256) {
    const int b = i >> 6, j = i & 63;
    float acc = bp[j];
    const float* e = ef + b * 64;
    for (int k = 0; k < 64; ++k) acc += e[k] * Wp[k * 64 + j];
    proj[i] = acc;
  }
  __syncthreads();
  for (int i = t; i < 512; i += 256) {
    const int b = i >> 4, p = i & 15;
    float acc = 0.f;
    const float* pr = proj + b * 64;
    const float* pt = prot + p * 64;
    for (int k = 0; k < 64; ++k) { const float dd = pr[k] - pt[k]; acc += dd * dd; }
    d2s[i] = acc;
  }
  __syncthreads();
  if (t < 32) {
    const float temp = fminf(fmaxf(__expf(log_temp[0]), 0.1f), 10.f);
    float z[8], m = -1e30f;
    for (int e = 0; e < 8; ++e) {
      const float d2pe = fminf(d2s[t * 16 + 2 * e], d2s[t * 16 + 2 * e + 1]);
      z[e] = (lg[t * 8 + e] - d2pe) / temp;
      m = fmaxf(m, z[e]);
    }
    float s = 0.f;
    for (int e = 0; e < 8; ++e) { z[e] = __expf(z[e] - m); s += z[e]; }
    for (int e = 0; e < 8; ++e) gw[t * 8 + e] = z[e] / s;
  }
}

__global__ void k_zero(float* __restrict__ y) {
  y[blockIdx.x * 256 + threadIdx.x] = 0.f;
}

// ---------------------------------------------------------------------------
// K5 (per expert): h = x[16384,32] @ Win_e[32,256] + b_in_e  (WMMA f32)
// A tile (16x32, fully contiguous rows) cooperatively staged in LDS with a
// padded row stride (36) so A-fragment lanes hit distinct banks.
// grid = 1024 mtiles * 2 ngroups = 2048, block = 256 (8 waves, 8 ntiles each)
// ---------------------------------------------------------------------------
__global__ void k_hgemm(const float* __restrict__ x, const float* __restrict__ Win,
                        const float* __restrict__ bin, int e, float* __restrict__ h) {
  __shared__ float Atile[16 * 36];
  const int bid = blockIdx.x;
  const int mtile = bid >> 1, ng = bid & 1;
  const int m0 = mtile * 16;
  const int w = threadIdx.x >> 5;
  const int n0 = (ng * 8 + w) * 16;

  const float* src = x + (size_t)m0 * 32;           // 512 contiguous floats
  for (int i = threadIdx.x; i < 512; i += 256) {
    Atile[(i >> 5) * 36 + (i & 31)] = src[i];
  }
  __syncthreads();

  const float* Be = Win + (size_t)e * 32 * 256;
  const v8f c = gemm_tile_f32_ldsA(Atile, 36, Be, 256, 32, n0);
  const int lane = threadIdx.x & 31, lm = lane & 15, hi = lane >> 4;
  const float* bb = bin + e * 256;
  for (int r = 0; r < 8; ++r) {
    const int row = m0 + r + hi * 8, col = n0 + lm;
    h[(size_t)row * 256 + col] = c[r] + bb[col];
  }
}

// ---------------------------------------------------------------------------
// K6 (per expert): fmag[b,:,ch] = |FFT512(h[b,:,ch])|. grid=8192, block=256
// ---------------------------------------------------------------------------
__global__ void k_fft(const float* __restrict__ h, float* __restrict__ fmag) {
  __shared__ float re[512], im[512];
  const int t = threadIdx.x;
  const int bc = blockIdx.x;
  const int b = bc >> 8, ch = bc & 255;
  const float* base = h + (size_t)b * 512 * 256 + ch;
  const float v0 = base[(size_t)t * 256];
  const float v1 = base[(size_t)(t + 256) * 256];
  const int j0 = (int)(__brev((unsigned)t) >> 23);
  const int j1 = (int)(__brev((unsigned)(t + 256)) >> 23);
  re[j0] = v0; im[j0] = 0.f;
  re[j1] = v1; im[j1] = 0.f;
  fft512(re, im, t);
  float* ob = fmag + (size_t)b * 512 * 256 + ch;
  ob[(size_t)t * 256]         = sqrtf(re[t] * re[t] + im[t] * im[t]);
  ob[(size_t)(t + 256) * 256] = sqrtf(re[t + 256] * re[t + 256] + im[t + 256] * im[t + 256]);
}

// ---------------------------------------------------------------------------
// K7 (per expert): u = relu(h + fmag@Wfft_e + b_fft_e), in-place into h.
// A tile (16x256 fp32, 16 KB) DMA'd into LDS by the Tensor Data Mover with
// LDS padding (4 DWORDs every 256 DWORDs -> row stride 260 floats, bank-
// conflict-free A fragments). Wave 0 issues tensor_load_to_lds, waits on
// TENSORcnt, then the workgroup barrier releases all 8 waves.
// grid = 1024 mtiles * 2 ngroups = 2048, block = 256
// ---------------------------------------------------------------------------
__global__ void k_fgemm(const float* __restrict__ fmag, const float* __restrict__ Wfft,
                        const float* __restrict__ bfft, int e, float* __restrict__ h) {
  __shared__ float Atile[16 * 260];
  const int bid = blockIdx.x;
  const int mtile = bid >> 1, ng = bid & 1;
  const int m0 = mtile * 16;
  const int w = threadIdx.x >> 5;
  const int n0 = (ng * 8 + w) * 16;

  if (w == 0) {
    // D# group 0: count=1 | lds_addr | global_addr[56:0] | type=2
    const unsigned long long ga =
        (unsigned long long)(uintptr_t)(fmag + (size_t)m0 * 256);
    u32x4 g0;
    g0.x = 1u;                                            // count=1 (valid D#)
    g0.y = (unsigned)(uintptr_t)&Atile[0];                // LDS byte address
    g0.z = (unsigned)(ga & 0xFFFFFFFFull);                // global_addr[31:0]
    g0.w = (unsigned)((ga >> 32) & 0x1FFFFFFull) | (2u << 30); // [56:32]|type=2
    // D# group 1: data_size=4B, pad_enable, pad_interval=256DW, pad_amount=4DW
    // tensor_dim0=256, tensor_dim1=16384, tile_dim0=256, tile_dim1=16,
    // tensor_dim0_stride=256
    i32x8 g1;
    g1[0] = (int)((2u << 16) | (1u << 20) | (7u << 22) | (3u << 25));
    g1[1] = (int)(256u << 16);     // tensor_dim0[15:0] in bits [63:48]
    g1[2] = (int)(16384u << 16);   // dim0 hi=0; tensor_dim1[15:0] in [95:80]
    g1[3] = (int)(256u << 16);     // dim1 hi=0; tile_dim0=256 in [127:112]
    g1[4] = 16;                    // tile_dim1=16; tile_dim2=0
    g1[5] = 256;                   // tensor_dim0_stride[31:0]
    g1[6] = 0;
    g1[7] = 0;
    i32x4 z4 = {0, 0, 0, 0};
#if __clang_major__ >= 23
    i32x8 z8 = {0, 0, 0, 0, 0, 0, 0, 0};
    __builtin_amdgcn_tensor_load_to_lds(g0, g1, z4, z4, z8, 0);
#else
    __builtin_amdgcn_tensor_load_to_lds(g0, g1, z4, z4, 0);
#endif
    __builtin_amdgcn_s_wait_tensorcnt(0);
  }
  __syncthreads();

  const float* Be = Wfft + (size_t)e * 256 * 256;
  const v8f c = gemm_tile_f32_ldsA(Atile, 260, Be, 256, 256, n0);
  const int lane = threadIdx.x & 31, lm = lane & 15, hi = lane >> 4;
  const float* bb = bfft + e * 256;
  for (int r = 0; r < 8; ++r) {
    const size_t idx = (size_t)(m0 + r + hi * 8) * 256 + n0 + lm;
    h[idx] = fmaxf(h[idx] + c[r] + bb[n0 + lm], 0.f);
  }
}

// ---------------------------------------------------------------------------
// K8 (per expert): decay scan s = a*s + (1-a)*u[t] over L=512.
// grid=B=32, block=H=256, coalesced stride-H loads.
// ---------------------------------------------------------------------------
__global__ void k_scan(const float* __restrict__ u, const float* __restrict__ alog,
                       int e, float* __restrict__ slast) {
  const int b = blockIdx.x, ch = threadIdx.x;
  const float a = 1.f / (1.f + __expf(-alog[e * 256 + ch]));
  const float om = 1.f - a;
  float s = 0.f;
  const float* base = u + (size_t)b * 512 * 256 + ch;
  for (int tt = 0; tt < 512; ++tt) s = a * s + om * base[(size_t)tt * 256];
  slast[b * 256 + ch] = s;
}

// ---------------------------------------------------------------------------
// K9 (per expert): y += gate_w[b,e]*(slast[32,256]@Whead_e[256,3072]+b_head_e)
// 2 Mtiles * 192 Ntiles = 384 waves -> grid=48, block=256 (WMMA f32)
// ---------------------------------------------------------------------------
__global__ void k_head(const float* __restrict__ slast, const float* __restrict__ Whead,
                       const float* __restrict__ bhead, const float* __restrict__ gw,
                       int e, float* __restrict__ y) {
  const int wid = blockIdx.x * 8 + (threadIdx.x >> 5);
  const int m0 = (wid / 192) * 16, n0 = (wid % 192) * 16;
  const float* Be = Whead + (size_t)e * 256 * 3072;
  const v8f c = gemm_tile_f32(slast, 256, Be, 3072, 256, m0, n0);
  const int lane = threadIdx.x & 31, lm = lane & 15, hi = lane >> 4;
  const float* bb = bhead + (size_t)e * 3072;
  for (int r = 0; r < 8; ++r) {
    const int row = m0 + r + hi * 8, col = n0 + lm;
    const float wgt = gw[row * 8 + e];
    const size_t idx = (size_t)row * 3072 + col;
    y[idx] += wgt * (c[r] + bb[col]);
  }
}

// ---------------------------------------------------------------------------
// Host launcher
// ---------------------------------------------------------------------------
extern "C" void kernel_launch(void* const* d_in, const int* in_sizes, int n_in,
                              void* d_out, int out_size, void* d_ws, size_t ws_size,
                              hipStream_t stream) {
  (void)in_sizes; (void)n_in; (void)out_size; (void)ws_size;
  const float* x     = (const float*)d_in[0];
  const float* prot  = (const float*)d_in[1];
  const float* Wp    = (const float*)d_in[2];
  const float* bp    = (const float*)d_in[3];
  const float* Wf1   = (const float*)d_in[4];
  const float* bf1   = (const float*)d_in[5];
  const float* Wf2   = (const float*)d_in[6];
  const float* bf2   = (const float*)d_in[7];
  const float* lng   = (const float*)d_in[8];
  const float* lnb   = (const float*)d_in[9];
  const float* W1    = (const float*)d_in[10];
  const float* b1    = (const float*)d_in[11];
  const float* W2    = (const float*)d_in[12];
  const float* b2    = (const float*)d_in[13];
  const float* Wg    = (const float*)d_in[14];
  const float* bg    = (const float*)d_in[15];
  const float* ltemp = (const float*)d_in[16];
  const float* Win   = (const float*)d_in[17];
  const float* bin   = (const float*)d_in[18];
  const float* Wfft  = (const float*)d_in[19];
  const float* bfft  = (const float*)d_in[20];
  const float* alog  = (const float*)d_in[21];
  const float* Whead = (const float*)d_in[22];
  const float* bhead = (const float*)d_in[23];
  float* y  = (float*)d_out;
  float* ws = (float*)d_ws;

  // workspace layout (float offsets)
  float* cf     = ws;                       // 32*224
  float* h1     = ws + 7168;                // 32*256
  float* efpre  = ws + 15360;               // 32*64
  float* ef     = ws + 17408;               // 32*64
  float* hidden = ws + 19456;               // 32*256
  float* latent = ws + 27648;               // 32*256
  float* lg     = ws + 35840;               // 32*8
  float* gw     = ws + 36096;               // 32*8
  float* slast  = ws + 36352;               // 32*256
  float* hbuf   = ws + 65536;               // 32*512*256 = 4,194,304
  float* fmag   = ws + 65536 + 4194304;     // 32*512*256

  k_stats  <<<1024, 256, 0, stream>>>(x, cf);
  k_efmlp  <<<1,    256, 0, stream>>>(cf, Wf1, bf1, Wf2, bf2, lng, lnb, h1, efpre, ef);
  k_latgemm<<<4,    256, 0, stream>>>(x, W1, b1, hidden);
  k_latent2<<<1,    256, 0, stream>>>(hidden, W2, b2, Wg, bg, latent, lg);
  k_gate   <<<1,    256, 0, stream>>>(ef, Wp, bp, prot, lg, ltemp, gw);
  k_zero   <<<384,  256, 0, stream>>>(y);

  for (int e = 0; e < 8; ++e) {
    k_hgemm<<<2048, 256, 0, stream>>>(x, Win, bin, e, hbuf);
    k_fft  <<<8192, 256, 0, stream>>>(hbuf, fmag);
    k_fgemm<<<2048, 256, 0, stream>>>(fmag, Wfft, bfft, e, hbuf);
    k_scan <<<32,   256, 0, stream>>>(hbuf, alog, e, slast);
    k_head <<<48,   256, 0, stream>>>(slast, Whead, bhead, gw, e, y);
  }
}